// LightGCN_87600152969701
// MI455X (gfx1250) — compile-verified
//
#include <hip/hip_runtime.h>

#define N_USERS 100000
#define N_ITEMS 200000
#define N_NODES (N_USERS + N_ITEMS)
#define DIM 64
#define NVEC (N_NODES * (DIM / 4))   // float4 elements = 4,800,000

// CDNA5 async gather-to-LDS path (gfx1250 only, guarded so compile never breaks)
#if defined(__gfx1250__) && __has_builtin(__builtin_amdgcn_global_load_async_to_lds_b128) && __has_builtin(__builtin_amdgcn_s_wait_asynccnt)
#define HAVE_ASYNC_LDS 1
typedef int v4i_t __attribute__((ext_vector_type(4)));
typedef __attribute__((address_space(1))) v4i_t* gvec_ptr;   // global int4*
typedef __attribute__((address_space(3))) v4i_t* lvec_ptr;   // LDS int4*
#else
#define HAVE_ASYNC_LDS 0
#endif

// Non-temporal helpers: keep streaming traffic (edge lists, accumulator) from
// evicting the L2-resident embedding tables x/y (76.8 MB each vs 192 MB L2).
__device__ __forceinline__ int   nt_load_i(const int* p)   { return __builtin_nontemporal_load(p); }
__device__ __forceinline__ float nt_load_f(const float* p) { return __builtin_nontemporal_load(p); }
__device__ __forceinline__ float4 nt_load_f4(const float4* p) {
    float4 v;
    v.x = __builtin_nontemporal_load(&p->x);
    v.y = __builtin_nontemporal_load(&p->y);
    v.z = __builtin_nontemporal_load(&p->z);
    v.w = __builtin_nontemporal_load(&p->w);
    return v;
}
__device__ __forceinline__ void nt_store_f4(float4* p, float4 v) {
    __builtin_nontemporal_store(v.x, &p->x);
    __builtin_nontemporal_store(v.y, &p->y);
    __builtin_nontemporal_store(v.z, &p->z);
    __builtin_nontemporal_store(v.w, &p->w);
}

// ---------------------------------------------------------------------------
// init: x = concat(user_emb, item_emb); out = x (acc starts at layer-0 emb);
//       y = 0 (scatter target for layer 1)
// ---------------------------------------------------------------------------
__global__ __launch_bounds__(256) void lgcn_init_kernel(
    const float* __restrict__ ue, const float* __restrict__ ie,
    float* __restrict__ x, float* __restrict__ y, float* __restrict__ out,
    int nvec)
{
    long i = (long)blockIdx.x * 256 + threadIdx.x;
    if (i >= nvec) return;
    const long uvec = (long)N_USERS * (DIM / 4);
    float4 v = (i < uvec) ? ((const float4*)ue)[i]
                          : ((const float4*)ie)[i - uvec];
    ((float4*)x)[i] = v;                         // resident: gather source
    ((float4*)y)[i] = make_float4(0.f, 0.f, 0.f, 0.f);  // resident: scatter target
    nt_store_f4(((float4*)out) + i, v);          // streaming accumulator
}

// ---------------------------------------------------------------------------
// SpMM scatter: y[rows[e],:] += vals[e] * x[cols[e],:]
// 16 lanes per edge, float4 per lane. Gather is double-buffered through LDS
// via GLOBAL_LOAD_ASYNC_TO_LDS_B128 (ASYNCcnt-tracked) so the DMA for edge
// e+G overlaps the atomic scatter of edge e.
// ---------------------------------------------------------------------------
__global__ __launch_bounds__(256) void lgcn_spmm_kernel(
    const int* __restrict__ rows, const int* __restrict__ cols,
    const float* __restrict__ vals, const float* __restrict__ x,
    float* __restrict__ y, int nnz)
{
    const int lane16 = threadIdx.x & 15;
    const long slot  = (long)blockIdx.x * 256 + threadIdx.x;
    const long g     = slot >> 4;                         // edge group id
    const long G     = ((long)gridDim.x * 256) >> 4;      // total groups

#if HAVE_ASYNC_LDS
    __shared__ float4 tile[2][256];                       // 8 KB double buffer
    int buf = 0;
    long e = g;
    if (e < nnz) {
        int c = nt_load_i(cols + e);
        const float4* src = (const float4*)(x + (long)c * DIM) + lane16;
        __builtin_amdgcn_global_load_async_to_lds_b128(
            (gvec_ptr)src, (lvec_ptr)&tile[0][threadIdx.x], 0, 0);
    }
    for (; e < nnz; e += G) {
        long en = e + G;
        if (en < nnz) {
            int cn = nt_load_i(cols + en);
            const float4* src = (const float4*)(x + (long)cn * DIM) + lane16;
            __builtin_amdgcn_global_load_async_to_lds_b128(
                (gvec_ptr)src, (lvec_ptr)&tile[buf ^ 1][threadIdx.x], 0, 0);
            __builtin_amdgcn_s_wait_asynccnt(1);   // current edge's tile ready
        } else {
            __builtin_amdgcn_s_wait_asynccnt(0);
        }
        float4 v = tile[buf][threadIdx.x];          // ds_load_b128
        float  a = nt_load_f(vals + e);
        long   r = nt_load_i(rows + e);
        float* dst = y + r * DIM + lane16 * 4;
        atomicAdd(dst + 0, v.x * a);
        atomicAdd(dst + 1, v.y * a);
        atomicAdd(dst + 2, v.z * a);
        atomicAdd(dst + 3, v.w * a);
        buf ^= 1;
    }
#else
    for (long e = g; e < nnz; e += G) {
        int   c = nt_load_i(cols + e);
        float4 v = ((const float4*)(x + (long)c * DIM))[lane16];
        float  a = nt_load_f(vals + e);
        long   r = nt_load_i(rows + e);
        float* dst = y + r * DIM + lane16 * 4;
        atomicAdd(dst + 0, v.x * a);
        atomicAdd(dst + 1, v.y * a);
        atomicAdd(dst + 2, v.z * a);
        atomicAdd(dst + 3, v.w * a);
    }
#endif
}

// ---------------------------------------------------------------------------
// accumulate: out = (out + y) * scale; optionally zero zbuf (becomes next y)
// out is streamed (NT); y stays resident (it becomes the next gather source).
// ---------------------------------------------------------------------------
__global__ __launch_bounds__(256) void lgcn_acc_kernel(
    float* __restrict__ out, const float* __restrict__ y,
    float* __restrict__ zbuf, float scale, int nvec)
{
    long i = (long)blockIdx.x * 256 + threadIdx.x;
    if (i >= nvec) return;
    float4 o = nt_load_f4(((const float4*)out) + i);
    float4 v = ((const float4*)y)[i];
    o.x = (o.x + v.x) * scale;
    o.y = (o.y + v.y) * scale;
    o.z = (o.z + v.z) * scale;
    o.w = (o.w + v.w) * scale;
    nt_store_f4(((float4*)out) + i, o);
    if (zbuf) ((float4*)zbuf)[i] = make_float4(0.f, 0.f, 0.f, 0.f);
}

// ---------------------------------------------------------------------------
extern "C" void kernel_launch(void* const* d_in, const int* in_sizes, int n_in,
                              void* d_out, int out_size, void* d_ws, size_t ws_size,
                              hipStream_t stream) {
    const float* ue   = (const float*)d_in[0];
    const float* ie   = (const float*)d_in[1];
    const int*   rows = (const int*)d_in[2];
    const int*   cols = (const int*)d_in[3];
    const float* vals = (const float*)d_in[4];
    const int    nnz  = in_sizes[2];

    float* out = (float*)d_out;                        // accumulator, node order
    float* x   = (float*)d_ws;                         // 76.8 MB
    float* y   = x + (size_t)N_NODES * DIM;            // 76.8 MB

    const int nvec    = NVEC;
    const int eblocks = (nvec + 255) / 256;
    const int sblocks = 16384;                         // grid-stride, ~24 edges/group

    lgcn_init_kernel<<<eblocks, 256, 0, stream>>>(ue, ie, x, y, out, nvec);

    for (int l = 0; l < 3; ++l) {
        lgcn_spmm_kernel<<<sblocks, 256, 0, stream>>>(rows, cols, vals, x, y, nnz);
        const float scale = (l == 2) ? 0.25f : 1.0f;   // fold /(L+1) into last pass
        float* zb = (l == 2) ? nullptr : x;            // old x becomes next y (zeroed)
        lgcn_acc_kernel<<<eblocks, 256, 0, stream>>>(out, y, zb, scale, nvec);
        float* t = x; x = y; y = t;                    // ping-pong
    }
}